// FairMOTLoss_48241072669138
// MI455X (gfx1250) — compile-verified
//
#include <hip/hip_runtime.h>
#include <hip/hip_bf16.h>
#include <math.h>

#define B_      8
#define N_      64
#define H_      152
#define W_      272
#define IMG_H_  608
#define IMG_W_  1088
#define EMB_    128
#define NC_     500
#define NPTS    (B_ * N_)     /* 512 */
#define NPAD    512           /* classes padded to multiple of 16 */
#define EPS_    1.0e-4f

typedef __attribute__((ext_vector_type(2))) float v2f;
typedef __attribute__((ext_vector_type(8))) float v8f;

/* ---- workspace layout (bytes) ---- */
#define OFF_ACC     0                               /* 16 floats of accumulators      */
#define OFF_IDX     1024                            /* ints: rr[512] cc[512] lab[512] */
#define OFF_MASK    8192                            /* B*H*W floats                   */
#define MASK_BYTES  (B_ * H_ * W_ * 4)              /* 1,323,008                      */
#define OFF_FEATS   (OFF_MASK + MASK_BYTES)         /* 512*128 floats                 */
#define FEATS_BYTES (NPTS * EMB_ * 4)
#define OFF_LOGITS  (OFF_FEATS + FEATS_BYTES)       /* 512*500 floats                 */

/* acc[0]=bs_loss acc[1]=off_loss acc[2]=pos_loss acc[3]=neg_loss acc[4]=npos acc[5]=ce_sum */

__global__ void fm_points(const float* __restrict__ gt,
                          const float* __restrict__ bs_head,
                          const float* __restrict__ off_head,
                          float* __restrict__ acc,
                          int* __restrict__ idx,
                          float* __restrict__ mask)
{
    int i = blockIdx.x * blockDim.x + threadIdx.x;
    if (i >= NPTS) return;
    int b = i / N_;
    const float* g = gt + (size_t)i * 5;
    float tid = g[0], xc = g[1], yc = g[2], bw = g[3], bh = g[4];
    float row_f = yc * (float)H_, col_f = xc * (float)W_;
    float fr = floorf(row_f), fc = floorf(col_f);
    int rr = (int)fr; rr = rr < 0 ? 0 : (rr > H_ - 1 ? H_ - 1 : rr);
    int cc = (int)fc; cc = cc < 0 ? 0 : (cc > W_ - 1 ? W_ - 1 : cc);
    mask[((size_t)b * H_ + rr) * W_ + cc] = 1.0f;     /* positions distinct per batch */
    idx[i] = rr;
    idx[NPTS + i] = cc;
    idx[2 * NPTS + i] = (tid > (float)NC_) ? (NC_ - 1) : ((int)tid - 1);

    size_t p0 = (((size_t)b * 2 + 0) * H_ + rr) * W_ + cc;
    size_t p1 = (((size_t)b * 2 + 1) * H_ + rr) * W_ + cc;
    float bs  = fabsf(bs_head[p0]  - bh * (float)IMG_H_) +
                fabsf(bs_head[p1]  - bw * (float)IMG_W_);
    float off = fabsf(off_head[p0] - (row_f - fr)) +
                fabsf(off_head[p1] - (col_f - fc));
    atomicAdd(&acc[0], bs);
    atomicAdd(&acc[1], off);
}

__global__ void fm_gather(const float* __restrict__ reid,
                          const int* __restrict__ idx,
                          float* __restrict__ feats)
{
    int t = blockIdx.x * blockDim.x + threadIdx.x;   /* 512*128 threads */
    int i = t >> 7, k = t & 127;
    int b = i >> 6;                                  /* i / N_ */
    int rr = idx[i], cc = idx[NPTS + i];
    feats[(size_t)i * EMB_ + k] =
        reid[(((size_t)b * EMB_ + k) * H_ + rr) * W_ + cc];
}

__global__ void fm_heatmap(const float* __restrict__ hm,
                           const float* __restrict__ mask,
                           float* __restrict__ acc)
{
    int t = blockIdx.x * blockDim.x + threadIdx.x;
    float pl = 0.f, nl = 0.f, np = 0.f;
    if (t < B_ * H_ * W_) {
        int c = t % W_;
        int r = (t / W_) % H_;
        int b = t / (W_ * H_);
        float gk[5];
        #pragma unroll
        for (int q = 0; q < 5; ++q) {
            float d = (float)(q - 2) * (1.0f / 3.0f);
            gk[q] = expf(-0.5f * d * d);
        }
        float gt_hm = 0.f;
        #pragma unroll
        for (int dr = -2; dr <= 2; ++dr) {
            int rn = r + dr;
            if (rn < 0 || rn >= H_) continue;
            #pragma unroll
            for (int dc = -2; dc <= 2; ++dc) {
                int cn = c + dc;
                if (cn < 0 || cn >= W_) continue;
                gt_hm += mask[((size_t)b * H_ + rn) * W_ + cn] * (gk[dr + 2] * gk[dc + 2]);
            }
        }
        float x = hm[t];
        float p = 1.0f / (1.0f + expf(-x));
        p = fminf(fmaxf(p, EPS_), 1.0f - EPS_);
        if (gt_hm == 1.0f) {
            float om = 1.0f - p;
            pl = om * om * logf(p);
            np = 1.0f;
        } else if (gt_hm < 1.0f) {
            float og = 1.0f - gt_hm;
            float og2 = og * og;
            nl = og2 * og2 * p * p * logf(1.0f - p);
        }
    }
    __shared__ float s[3][256];
    int lt = threadIdx.x;
    s[0][lt] = pl; s[1][lt] = nl; s[2][lt] = np;
    __syncthreads();
    for (int st = 128; st > 0; st >>= 1) {
        if (lt < st) {
            s[0][lt] += s[0][lt + st];
            s[1][lt] += s[1][lt + st];
            s[2][lt] += s[2][lt + st];
        }
        __syncthreads();
    }
    if (lt == 0) {
        atomicAdd(&acc[2], s[0][0]);
        atomicAdd(&acc[3], s[1][0]);
        atomicAdd(&acc[4], s[2][0]);
    }
}

/* 512x500x128 f32 GEMM: logits = feats @ W_cls^T + b, via V_WMMA_F32_16X16X4_F32.
   One wave per 16x16 tile; K marched 4 at a time. N padded to 512 with a zeroed
   B fragment for cols >= 500 (multiplicative mask, no EXEC divergence). */
__global__ void fm_gemm_wmma(const float* __restrict__ feats,
                             const float* __restrict__ Wc,
                             const float* __restrict__ bc,
                             float* __restrict__ logits)
{
    int lane = threadIdx.x;               /* 32 threads = 1 wave */
    int m0 = blockIdx.x * 16;
    int n0 = blockIdx.y * 16;
    int lm = lane & 15;
    int kh = (lane >> 4) << 1;            /* lanes 0-15 -> K+0/+1, lanes 16-31 -> K+2/+3 */

    const float* arow = feats + (size_t)(m0 + lm) * EMB_;
    int ncol = n0 + lm;
    int nclamp = ncol < NC_ ? ncol : NC_ - 1;
    float nmask = ncol < NC_ ? 1.0f : 0.0f;
    const float* brow = Wc + (size_t)nclamp * EMB_;   /* B[k][n] = W_cls[n][k] */

    v8f c = {};
    for (int k0 = 0; k0 < EMB_; k0 += 4) {
        v2f a, b;
        a.x = arow[k0 + kh];
        a.y = arow[k0 + kh + 1];
        b.x = brow[k0 + kh] * nmask;
        b.y = brow[k0 + kh + 1] * nmask;
        c = __builtin_amdgcn_wmma_f32_16x16x4_f32(
                /*neg_a=*/false, a, /*neg_b=*/false, b,
                /*c_mod=*/(short)0, c, /*reuse_a=*/false, /*reuse_b=*/false);
    }

    /* D layout: VGPR r -> row m0+r (lanes 0-15) / m0+8+r (lanes 16-31), col = n0 + (lane&15) */
    int rbase = m0 + ((lane >> 4) << 3);
    int col = n0 + lm;
    if (col < NC_) {
        float bias = bc[col];
        #pragma unroll
        for (int r = 0; r < 8; ++r)
            logits[(size_t)(rbase + r) * NC_ + col] = c[r] + bias;
    }
}

__global__ void fm_ce(const float* __restrict__ logits,
                      const int* __restrict__ idx,
                      float* __restrict__ acc)
{
    int row = blockIdx.x;
    const float* lrow = logits + (size_t)row * NC_;
    int lt = threadIdx.x;                 /* 256 */
    __shared__ float s[256];

    float mx = -3.0e38f;
    for (int j = lt; j < NC_; j += 256) mx = fmaxf(mx, lrow[j]);
    s[lt] = mx; __syncthreads();
    for (int st = 128; st > 0; st >>= 1) {
        if (lt < st) s[lt] = fmaxf(s[lt], s[lt + st]);
        __syncthreads();
    }
    mx = s[0]; __syncthreads();

    float sum = 0.f;
    for (int j = lt; j < NC_; j += 256) sum += expf(lrow[j] - mx);
    s[lt] = sum; __syncthreads();
    for (int st = 128; st > 0; st >>= 1) {
        if (lt < st) s[lt] += s[lt + st];
        __syncthreads();
    }
    if (lt == 0) {
        int lab = idx[2 * NPTS + row];
        float ce = logf(s[0]) + mx - lrow[lab];
        atomicAdd(&acc[5], ce);
    }
}

__global__ void fm_final(const float* __restrict__ acc, float* __restrict__ out)
{
    if (threadIdx.x == 0 && blockIdx.x == 0) {
        float bs = acc[0], off = acc[1], pl = acc[2], nl = acc[3];
        float np = acc[4], ces = acc[5];
        float hm   = (np >= 1.0f) ? (-(pl + nl) / np) : (-nl);
        float reid = ces / (float)N_;     /* mean over N, sum over B */
        float det  = hm + 0.1f * bs + off;
        out[0] = bs;
        out[1] = off;
        out[2] = det;
        out[3] = hm;
        out[4] = reid;
        out[5] = det + reid;
    }
}

extern "C" void kernel_launch(void* const* d_in, const int* in_sizes, int n_in,
                              void* d_out, int out_size, void* d_ws, size_t ws_size,
                              hipStream_t stream)
{
    (void)in_sizes; (void)n_in; (void)out_size; (void)ws_size;
    const float* hm   = (const float*)d_in[0];
    const float* bsh  = (const float*)d_in[1];
    const float* offh = (const float*)d_in[2];
    const float* reid = (const float*)d_in[3];
    const float* gt   = (const float*)d_in[4];
    const float* Wc   = (const float*)d_in[5];
    const float* bc   = (const float*)d_in[6];

    char*  ws     = (char*)d_ws;
    float* acc    = (float*)(ws + OFF_ACC);
    int*   idx    = (int*)(ws + OFF_IDX);
    float* mask   = (float*)(ws + OFF_MASK);
    float* feats  = (float*)(ws + OFF_FEATS);
    float* logits = (float*)(ws + OFF_LOGITS);

    /* zero accumulators + mask every call (graph-replay safe, capturable) */
    hipMemsetAsync(ws, 0, (size_t)OFF_MASK + MASK_BYTES, stream);

    fm_points<<<(NPTS + 255) / 256, 256, 0, stream>>>(gt, bsh, offh, acc, idx, mask);
    fm_gather<<<(NPTS * EMB_) / 256, 256, 0, stream>>>(reid, idx, feats);
    int tot = B_ * H_ * W_;
    fm_heatmap<<<(tot + 255) / 256, 256, 0, stream>>>(hm, mask, acc);
    fm_gemm_wmma<<<dim3(NPTS / 16, NPAD / 16), 32, 0, stream>>>(feats, Wc, bc, logits);
    fm_ce<<<NPTS, 256, 0, stream>>>(logits, idx, acc);
    fm_final<<<1, 32, 0, stream>>>(acc, (float*)d_out);
}